// DeepSeekMoELayer_75101798138194
// MI455X (gfx1250) — compile-verified
//
#include <hip/hip_runtime.h>
#include <math.h>

typedef __attribute__((ext_vector_type(2))) float v2f;
typedef __attribute__((ext_vector_type(8))) float v8f;

constexpr int Hdim = 768;
constexpr int Idim = 3072;
constexpr int NEXP = 16;    // routed experts
constexpr int NSH  = 2;     // shared experts
constexpr int NTOK = 2048;  // 2*1024
constexpr int CHUNK = 64;   // I-chunk per iteration
constexpr float SCALE = 1.0f;

// ---------------------------------------------------------------------------
// Kernel 1: router. One wave (32 lanes) per token; lanes 0-15 own experts.
// Writes per-token top-2 (idx, weight) pairs deterministically.
// ---------------------------------------------------------------------------
__global__ __launch_bounds__(128) void moe_router(
    const float* __restrict__ x, const float* __restrict__ gate_w,
    int* __restrict__ sel_idx, float* __restrict__ sel_w) {
  const int wave = threadIdx.x >> 5;
  const int lane = threadIdx.x & 31;
  const int n = blockIdx.x * 4 + wave;

  float logit = -INFINITY;
  if (lane < NEXP) {
    const float* gw = gate_w + lane * Hdim;
    const float* xr = x + (size_t)n * Hdim;
    float s = 0.f;
    for (int h = 0; h < Hdim; ++h) s = fmaf(xr[h], gw[h], s);
    logit = s;
  }
  // softmax over lanes 0..15 (xor offsets <=8 stay inside the 16-lane half)
  float mx = logit;
  for (int off = 8; off; off >>= 1) mx = fmaxf(mx, __shfl_xor(mx, off));
  float p = (lane < NEXP) ? __expf(logit - mx) : 0.f;
  float sum = p;
  for (int off = 8; off; off >>= 1) sum += __shfl_xor(sum, off);
  float prob = p / sum;
  // top-1 (lower index wins ties)
  float v1 = prob; int i1 = lane;
  for (int off = 8; off; off >>= 1) {
    float ov = __shfl_xor(v1, off); int oi = __shfl_xor(i1, off);
    if (ov > v1 || (ov == v1 && oi < i1)) { v1 = ov; i1 = oi; }
  }
  // top-2
  float v2 = (lane == i1) ? -1.f : prob; int i2 = lane;
  for (int off = 8; off; off >>= 1) {
    float ov = __shfl_xor(v2, off); int oi = __shfl_xor(i2, off);
    if (ov > v2 || (ov == v2 && oi < i2)) { v2 = ov; i2 = oi; }
  }
  const float s12 = v1 + v2 + 1e-8f;
  if (lane == 0) {
    sel_idx[n * 2 + 0] = i1;
    sel_idx[n * 2 + 1] = i2;
    sel_w[n * 2 + 0] = v1 / s12 * SCALE;
    sel_w[n * 2 + 1] = v2 / s12 * SCALE;
  }
}

// ---------------------------------------------------------------------------
// Kernel 2: deterministic compaction. One wave per expert scans tokens in
// order, building per-expert token lists via ballot prefix sums.
// ---------------------------------------------------------------------------
__global__ __launch_bounds__(32) void moe_compact(
    const int* __restrict__ sel_idx, const float* __restrict__ sel_w,
    int* __restrict__ counts, int* __restrict__ token_ids,
    float* __restrict__ token_w) {
  const int e = blockIdx.x;
  const int lane = threadIdx.x;
  int cnt = 0;
  for (int base = 0; base < NTOK; base += 32) {
    const int n = base + lane;
    const int i1 = sel_idx[n * 2], i2 = sel_idx[n * 2 + 1];
    const bool sel = (i1 == e) || (i2 == e);
    const float w = (i1 == e) ? sel_w[n * 2] : ((i2 == e) ? sel_w[n * 2 + 1] : 0.f);
    unsigned long long m = __ballot(sel);
    const unsigned long long lt = (1ull << lane) - 1ull;
    const int pos = cnt + __popcll(m & lt);
    if (sel) {
      token_ids[e * NTOK + pos] = n;
      token_w[e * NTOK + pos] = w;
    }
    cnt += __popcll(m);
  }
  if (lane == 0) counts[e] = cnt;
}

// ---------------------------------------------------------------------------
// Kernel 3: fused gather -> gate/up GEMM -> silu*up*w -> down GEMM -> atomic
// scatter. Block = 256 threads (8 waves), one 16-token tile of one expert.
// Experts 16,17 are shared experts (all tokens, weight 0.5 each).
// ---------------------------------------------------------------------------
__global__ __launch_bounds__(256) void moe_ffn(
    const float* __restrict__ x,
    const float* __restrict__ swg, const float* __restrict__ swu,
    const float* __restrict__ swd,
    const float* __restrict__ wg, const float* __restrict__ wu,
    const float* __restrict__ wd,
    const int* __restrict__ counts, const int* __restrict__ token_ids,
    const float* __restrict__ token_w, float* __restrict__ out) {
  __shared__ float xt[Hdim * 16];      // xt[h][m], k-major (48 KB)
  __shared__ float gbuf[CHUNK * 16];   // g[k][m] -> becomes h[k][m]
  __shared__ float ubuf[CHUNK * 16];   // u[k][m]
  __shared__ float s_w[16];
  __shared__ int s_tok[16];

  const int e = blockIdx.y;    // 0..17
  const int tile = blockIdx.x; // 0..127

  int cnt;
  const float *pwg, *pwu, *pwd;
  if (e < NEXP) {
    cnt = counts[e];
    pwg = wg + (size_t)e * Hdim * Idim;
    pwu = wu + (size_t)e * Hdim * Idim;
    pwd = wd + (size_t)e * Idim * Hdim;
  } else {
    cnt = NTOK;
    const int s = e - NEXP;
    pwg = swg + (size_t)s * Hdim * Idim;
    pwu = swu + (size_t)s * Hdim * Idim;
    pwd = swd + (size_t)s * Idim * Hdim;
  }
  if (tile * 16 >= cnt) return;
  const int numTok = (cnt - tile * 16 < 16) ? (cnt - tile * 16) : 16;

  const int t = threadIdx.x;
  if (t < 16) {
    if (t < numTok) {
      if (e < NEXP) {
        s_tok[t] = token_ids[e * NTOK + tile * 16 + t];
        s_w[t] = token_w[e * NTOK + tile * 16 + t];
      } else {
        s_tok[t] = tile * 16 + t;
        s_w[t] = 0.5f;  // mean over S=2 shared experts
      }
    } else {
      s_tok[t] = 0;
      s_w[t] = 0.f;
    }
  }
  __syncthreads();

  // gather 16 token rows into LDS, transposed to k-major
  {
    const int m = t >> 4, h0 = t & 15;
    const bool valid = m < numTok;
    const float* xr = x + (size_t)s_tok[m] * Hdim;
    for (int h = h0; h < Hdim; h += 16) xt[h * 16 + m] = valid ? xr[h] : 0.f;
  }
  __syncthreads();

  const int wave = t >> 5;
  const int lane = t & 31;
  const int lm = lane & 15;  // M (A/C rows) or N (B/C cols) sub-index
  const int lhi = lane >> 4; // upper-half-wave flag

  // persistent output accumulators: cols [wave*96, wave*96+96)
  v8f acc[6];
#pragma unroll
  for (int i = 0; i < 6; ++i) acc[i] = (v8f)0.f;

  for (int cb = 0; cb < Idim; cb += CHUNK) {
    // ---- GEMM1: 16x16 tile of G (waves 0-3) or U (waves 4-7), K = Hdim ----
    {
      const float* wbase = (wave < 4) ? pwg : pwu;
      const int col0 = cb + ((wave & 3) << 4);
      v8f c = (v8f)0.f;
      for (int k0 = 0; k0 < Hdim; k0 += 4) {
        const int bk = k0 + (lhi << 1);
        v2f a = {xt[bk * 16 + lm], xt[(bk + 1) * 16 + lm]};
        v2f b = {wbase[(size_t)bk * Idim + col0 + lm],
                 wbase[(size_t)(bk + 1) * Idim + col0 + lm]};
        c = __builtin_amdgcn_wmma_f32_16x16x4_f32(false, a, false, b,
                                                  (short)0, c, false, false);
      }
      float* dst = (wave < 4) ? gbuf : ubuf;
      const int ccol = ((wave & 3) << 4) + lm;  // chunk-local column 0..63
#pragma unroll
      for (int r = 0; r < 8; ++r) dst[ccol * 16 + (r + (lhi << 3))] = c[r];
    }
    __syncthreads();

    // ---- h = silu(g) * u * route_weight (weight folded in: linear wrt wd) --
    // sigmoid via v_rcp_f32 (~1 ulp) instead of the IEEE div fixup sequence.
    for (int i = t; i < CHUNK * 16; i += 256) {
      const float g = gbuf[i];
      const float u = ubuf[i];
      const float sig = __builtin_amdgcn_rcpf(1.f + __expf(-g));
      gbuf[i] = g * sig * u * s_w[i & 15];
    }
    __syncthreads();

    // ---- GEMM2: acc(16x96) += h(16xCHUNK) @ wd[cb:cb+CHUNK, wave cols] ----
    {
      const int colw = wave * 96;
      for (int k0 = 0; k0 < CHUNK; k0 += 4) {
        const int bk = k0 + (lhi << 1);
        v2f a = {gbuf[bk * 16 + lm], gbuf[(bk + 1) * 16 + lm]};
        const float* r0 = pwd + (size_t)(cb + bk) * Hdim;
        const float* r1 = pwd + (size_t)(cb + bk + 1) * Hdim;
#pragma unroll
        for (int tt = 0; tt < 6; ++tt) {
          const int col = colw + tt * 16 + lm;
          v2f b = {r0[col], r1[col]};
          acc[tt] = __builtin_amdgcn_wmma_f32_16x16x4_f32(
              false, a, false, b, (short)0, acc[tt], false, false);
        }
      }
    }
    __syncthreads();  // gbuf/ubuf reused next chunk
  }

  // ---- scatter: out[token, col] += acc (weights already folded into h) ----
  {
    const int colw = wave * 96;
#pragma unroll
    for (int tt = 0; tt < 6; ++tt) {
      const int col = colw + tt * 16 + lm;
#pragma unroll
      for (int r = 0; r < 8; ++r) {
        const int m = r + (lhi << 3);
        atomicAdd(out + (size_t)s_tok[m] * Hdim + col, acc[tt][r]);
      }
    }
  }
}

// ---------------------------------------------------------------------------
extern "C" void kernel_launch(void* const* d_in, const int* in_sizes, int n_in,
                              void* d_out, int out_size, void* d_ws,
                              size_t ws_size, hipStream_t stream) {
  const float* x      = (const float*)d_in[0];
  const float* gate_w = (const float*)d_in[1];
  const float* swg    = (const float*)d_in[2];
  const float* swu    = (const float*)d_in[3];
  const float* swd    = (const float*)d_in[4];
  const float* wg     = (const float*)d_in[5];
  const float* wu     = (const float*)d_in[6];
  const float* wd     = (const float*)d_in[7];
  float* out = (float*)d_out;

  char* ws = (char*)d_ws;
  int*   sel_idx   = (int*)(ws);                       // 2048*2 ints   (16 KB)
  float* sel_w     = (float*)(ws + 16384);             // 2048*2 floats (16 KB)
  int*   counts    = (int*)(ws + 32768);               // 16 ints
  int*   token_ids = (int*)(ws + 33024);               // 16*2048 ints (128 KB)
  float* token_w   = (float*)(ws + 33024 + 131072);    // 16*2048 floats

  hipMemsetAsync(d_out, 0, sizeof(float) * (size_t)NTOK * Hdim, stream);
  moe_router<<<NTOK / 4, 128, 0, stream>>>(x, gate_w, sel_idx, sel_w);
  moe_compact<<<NEXP, 32, 0, stream>>>(sel_idx, sel_w, counts, token_ids,
                                       token_w);
  dim3 grid(NTOK / 16, NEXP + NSH);
  moe_ffn<<<grid, 256, 0, stream>>>(x, swg, swu, swd, wg, wu, wd, counts,
                                    token_ids, token_w, out);
}